// NN_19902878450375
// MI455X (gfx1250) — compile-verified
//
#include <hip/hip_runtime.h>

typedef float v2f __attribute__((ext_vector_type(2)));
typedef float v8f __attribute__((ext_vector_type(8)));

#define BATCH 32768
#define NF    200
#define DIN   5
#define NH    15
#define NOUT  30

__device__ __forceinline__ v8f wmma_f32_k4(v2f a, v2f b, v8f c) {
  // D = A(16x4 f32) * B(4x16 f32) + C(16x16 f32)
  return __builtin_amdgcn_wmma_f32_16x16x4_f32(
      /*neg_a=*/false, a, /*neg_b=*/false, b,
      /*c_mod=*/(short)0, c, /*reuse_a=*/false, /*reuse_b=*/false);
}

// ds_swizzle group-of-32 encoding: [14:10]=xor  [9:5]=or  [4:0]=and
template <int IMM>
__device__ __forceinline__ float swz(float x) {
  return __int_as_float(__builtin_amdgcn_ds_swizzle(__float_as_int(x), IMM));
}
#define SWAP16 0x401f   // xor=0x10, and=0x1f : swap lane halves
#define XOR1   0x041f
#define XOR2   0x081f
#define XOR4   0x101f
#define XOR8   0x201f

__global__ __launch_bounds__(256) void mlp200_wmma_kernel(
    const float* __restrict__ y,  const float* __restrict__ W1,
    const float* __restrict__ b1, const float* __restrict__ W2,
    const float* __restrict__ b2, const float* __restrict__ W3,
    const float* __restrict__ b3, float* __restrict__ out)
{
  const int  lane = threadIdx.x & 31;
  const int  wave = threadIdx.x >> 5;
  const int  q    = lane >> 4;          // lane half
  const bool qhi  = (q != 0);
  const int  n    = lane & 15;
  const int  b0   = (blockIdx.x * 8 + wave) * 16;

  // loop-invariant, clamped per-lane offsets (all loads below are in-bounds)
  const int  nc   = (n < NH - 1) ? n : (NH - 1);        // min(n,14): W1/b1 column
  const int  oc2  = 16 + ((n < 14) ? n : 13);           // min(16+n,29): tile-1 col
  const bool o2ok = (n < 14);

  // 32-bit per-lane element indices against uniform (SGPR) bases -> GVS form
  const int  iy0  = (b0 + n) * (NF * DIN) + 2 * q;      // layer1 B chunk0 (K=2q)
  const int  iy1  = iy0 + 1;                            // K=2q+1
  const int  iy4  = (b0 + n) * (NF * DIN) + 4;          // i = 4

  const int iW1a = (2 * q) * NH + nc;   // W1[2q][h]
  const int iW1b = iW1a + NH;           // W1[2q+1][h]
  const int iW1c = 4 * NH + nc;         // W1[4][h]

  float acc[8];
  #pragma unroll
  for (int r = 0; r < 8; ++r) acc[r] = 0.f;

  #pragma unroll 2
  for (int f = 0; f < NF; ++f) {
    const float* yb  = y  + f * DIN;            // uniform bases, bumped per f
    const float* W1f = W1 + f * (DIN * NH);
    const float* W2f = W2 + f * (NH * NOUT);
    const float* b1f = b1 + f * NH;
    const float* b2f = b2 + f * NOUT;
    const float* W3f = W3 + f * NOUT;

    // ---- layer 1: D1 = W1^T (A: M=h,K=i) x y^T (B: K=i,N=b), bias via K=5 ----
    v8f c1 = {};
    {
      v2f a, bb;
      // K-chunk 0: lane supplies K = 2q, 2q+1 (all real data)
      a.x  = W1f[iW1a];
      a.y  = W1f[iW1b];
      bb.x = yb[iy0];
      bb.y = yb[iy1];
      c1 = wmma_f32_k4(a, bb, c1);
      // K-chunk 1: K=4 -> data, K=5 -> b1 (B row of ones), K=6,7 -> B rows zeroed
      a.x  = W1f[iW1c];                 // garbage A col 6 ok: B row 6 is zero
      a.y  = b1f[nc];                   // A col 5 = b1[h]; col 7 garbage ok (row 7 zero)
      float y4 = yb[iy4];
      bb.x = qhi ? 0.f : y4;
      bb.y = qhi ? 0.f : 1.f;
      c1 = wmma_f32_k4(a, bb, c1);
    }

    // ---- C-layout -> A-layout fixup: swap lane halves with ds_swizzle ----
    float hv[8], sw[8];
    #pragma unroll
    for (int r = 0; r < 8; ++r) { hv[r] = c1[r]; sw[r] = swz<SWAP16>(c1[r]); }

    v2f a0, a1, a2, a3;                 // A chunks K=0..3, 4..7, 8..11, 12..15
    a0.x = qhi ? sw[2] : hv[0];
    a0.y = qhi ? sw[3] : hv[1];
    a1.x = qhi ? sw[6] : hv[4];
    a1.y = qhi ? sw[7] : hv[5];
    a2.x = qhi ? hv[2] : sw[0];
    a2.y = qhi ? hv[3] : sw[1];
    a3.x = qhi ? hv[6] : sw[4];
    a3.y = qhi ? 1.f   : sw[5];         // A col 15 = 1.0 -> layer-2 bias row

    // ---- layer 2: D2 = h1(A) x W2(B), two N-tiles; bias via K=15 row = b2 ----
    v8f c2a = {}, c2b = {};
    #pragma unroll
    for (int k = 0; k < 4; ++k) {
      const v2f av = (k == 0) ? a0 : (k == 1) ? a1 : (k == 2) ? a2 : a3;
      const int h0  = 4 * k + 2 * q;                    // <= 14, always valid
      int       h1i = h0 + 1; if (h1i > NH - 1) h1i = NH - 1;   // clamp (k=3,q=1)
      v2f bta, btb;
      bta.x = W2f[h0  * NOUT + n];
      btb.x = W2f[h0  * NOUT + oc2];    // cols 30/31 clamped: dead D cols, finite
      float bay = W2f[h1i * NOUT + n];
      float bby = W2f[h1i * NOUT + oc2];
      if (k == 3) {                     // K=15 row carries b2 (q==1 lanes)
        float b2a = b2f[n];
        float b2b = b2f[oc2];
        bta.y = qhi ? b2a : bay;
        btb.y = qhi ? b2b : bby;
      } else {
        bta.y = bay;
        btb.y = bby;
      }
      c2a = wmma_f32_k4(av, bta, c2a);
      c2b = wmma_f32_k4(av, btb, c2b);
    }

    // ---- fused ReLU + dot(W3), accumulate over features ----
    float w3a = W3f[n];
    float w3t = W3f[oc2];
    float w3b = o2ok ? w3t : 0.f;       // kills the two pad columns of tile 1
    #pragma unroll
    for (int r = 0; r < 8; ++r) {
      float va = c2a[r] > 0.f ? c2a[r] : 0.f;
      float vb = c2b[r] > 0.f ? c2b[r] : 0.f;
      acc[r] += va * w3a + vb * w3b;
    }
  }

  // ---- butterfly-reduce the 16 lanes of each half ----
  #pragma unroll
  for (int r = 0; r < 8; ++r) {
    float v = acc[r];
    v += swz<XOR1>(v);
    v += swz<XOR2>(v);
    v += swz<XOR4>(v);
    v += swz<XOR8>(v);
    acc[r] = v;
  }

  if (n == 0) {                         // lanes 0 and 16 write 8 rows each
    const float b3v = b3[0];
    float* op = out + b0 + 8 * q;       // rows b0+8q .. b0+8q+7
    float4 o0 = make_float4(acc[0] + b3v, acc[1] + b3v, acc[2] + b3v, acc[3] + b3v);
    float4 o1 = make_float4(acc[4] + b3v, acc[5] + b3v, acc[6] + b3v, acc[7] + b3v);
    reinterpret_cast<float4*>(op)[0] = o0;
    reinterpret_cast<float4*>(op)[1] = o1;
  }
}

extern "C" void kernel_launch(void* const* d_in, const int* in_sizes, int n_in,
                              void* d_out, int out_size, void* d_ws, size_t ws_size,
                              hipStream_t stream) {
  (void)in_sizes; (void)n_in; (void)d_ws; (void)ws_size; (void)out_size;
  const float* y  = (const float*)d_in[0];
  const float* W1 = (const float*)d_in[1];
  const float* b1 = (const float*)d_in[2];
  const float* W2 = (const float*)d_in[3];
  const float* b2 = (const float*)d_in[4];
  const float* W3 = (const float*)d_in[5];
  const float* b3 = (const float*)d_in[6];
  float* out = (float*)d_out;

  dim3 grid(BATCH / (16 * 8));   // 256 blocks x 8 waves = 2048 tiles of 16 rows
  dim3 block(256);
  mlp200_wmma_kernel<<<grid, block, 0, stream>>>(y, W1, b1, W2, b2, W3, b3, out);
}